// DeepFM_53180285059513
// MI455X (gfx1250) — compile-verified
//
#include <hip/hip_runtime.h>
#include <hip/hip_bf16.h>
#include <cstdint>

typedef float v2f __attribute__((ext_vector_type(2)));
typedef float v8f __attribute__((ext_vector_type(8)));

constexpr int N_SAMP = 8192;
constexpr int F_FEAT = 45;
constexpr int E_DIM  = 256;
constexpr int DEEP   = F_FEAT * E_DIM;   // 11520
constexpr int LDA    = 260;              // padded LDS row stride (floats): 16B-aligned rows, conflict-free b64 reads
constexpr float BN_EPS = 1e-5f;

// PERM / FIELD_OFF baked from the reference's SECTIONS/FEATURE_SIZES
__constant__ int PERM_C[F_FEAT] = {
  0,1,2,3,4,5,6, 7,8,9, 10,11,12,13,14,15,16,17,18,19,20,21,22,23,
  26,27, 28,29, 24,25, 30,31,32,33,34,35,36,37,38,39,40,41,42,43,44 };
__constant__ int FOFF_C[F_FEAT] = {
  0,300,1300,2300,2800,2950,2962, 2993,2993,2993,
  3023,3023,3023,3023,3023,3023,3023,3023,3023,3023,3023,3023,3023,3023,
  3223,3223, 53223,53223, 103223,103223,
  123223,123223,123223,123223,123223,123223,123223,123223,
  123223,123223,123223,123223,123223,123223,123223 };

// ---------------------------------------------------------------------------
// K1: fused gather + FM(1st+2nd order) + layer-1 GEMM via V_WMMA_F32_16X16X4_F32
// Block = 256 threads (8 waves), handles 16 samples (one WMMA M-tile).
// Waves 0,1 own N-columns 0-15 / 16-31 of l1_w.
// ---------------------------------------------------------------------------
__global__ __launch_bounds__(256) void k1_gather_fm_gemm(
    const int* __restrict__ Xi, const float* __restrict__ Xv,
    const float* __restrict__ W1, const float* __restrict__ W2,
    const float* __restrict__ l1w, const float* __restrict__ l1b,
    float* __restrict__ z1, float* __restrict__ fm)
{
  __shared__ float Abuf[2][16 * LDA];
  __shared__ float red[256];

  const int t       = threadIdx.x;
  const int sampBase= blockIdx.x * 16;
  const int sLoc    = t >> 4;          // local sample 0..15 (group of 16 threads)
  const int lane16  = t & 15;          // lane within sample group
  const int n       = sampBase + sLoc;

  const int wave = t >> 5;             // 0..7
  const int lane = t & 31;
  const int jcol = (wave << 4) + (lane & 15);   // output column for WMMA waves
  const int krow = (lane >> 4) << 1;            // 0 (lanes 0-15) or 2 (lanes 16-31)

  float4 s4[4], q4[4];
  #pragma unroll
  for (int i = 0; i < 4; ++i) { s4[i] = make_float4(0,0,0,0); q4[i] = make_float4(0,0,0,0); }
  float fm1 = 0.0f;
  v8f acc = {};                        // WMMA C/D accumulator (waves 0,1)

  // ---- stage helper (feature f -> Abuf[buf]) ----
  auto stage = [&](int f, int buf) {
    const int col = PERM_C[f];
    const int idx = Xi[n * F_FEAT + col] + FOFF_C[f];
    const float xv = Xv[n * F_FEAT + col];
    const float* __restrict__ row = W2 + (size_t)idx * E_DIM;
    float* __restrict__ Ad = &Abuf[buf][sLoc * LDA];
    #pragma unroll
    for (int it = 0; it < 4; ++it) {
      const int k = it * 64 + lane16 * 4;
      float4 v = *reinterpret_cast<const float4*>(row + k);
      v.x *= xv; v.y *= xv; v.z *= xv; v.w *= xv;
      s4[it].x += v.x;     s4[it].y += v.y;     s4[it].z += v.z;     s4[it].w += v.w;
      q4[it].x += v.x*v.x; q4[it].y += v.y*v.y; q4[it].z += v.z*v.z; q4[it].w += v.w*v.w;
      *reinterpret_cast<float4*>(Ad + k) = v;
    }
    if (lane16 == 0) fm1 += W1[idx] * xv;
  };

  stage(0, 0);
  __syncthreads();

  for (int f = 1; f <= F_FEAT; ++f) {
    const int cbuf = (f - 1) & 1;
    if (wave < 2) {
      // B fragment base: column jcol of l1_w, K-rows krow..krow+1 within this feature chunk
      const float* __restrict__ Brow =
          l1w + (size_t)jcol * DEEP + (size_t)(f - 1) * E_DIM + krow;
      const float* __restrict__ Aptr = &Abuf[cbuf][(lane & 15) * LDA + krow];
      #pragma unroll 8
      for (int kk = 0; kk < E_DIM; kk += 4) {
        v2f a = *reinterpret_cast<const v2f*>(Aptr + kk);   // ds_load_b64
        v2f b = *reinterpret_cast<const v2f*>(Brow + kk);   // global_load_b64
        acc = __builtin_amdgcn_wmma_f32_16x16x4_f32(
            /*neg_a=*/false, a, /*neg_b=*/false, b,
            /*c_mod=*/(short)0, acc, /*reuse_a=*/false, /*reuse_b=*/false);
      }
    }
    if (f < F_FEAT) stage(f, f & 1);
    __syncthreads();
  }

  // ---- write z1 tile (C/D layout: VGPR r -> M = r + 8*(lane>=16), N = lane%16) ----
  if (wave < 2) {
    const float bj = l1b[jcol];
    #pragma unroll
    for (int r = 0; r < 8; ++r) {
      const int M = r + ((lane >> 4) << 3);
      z1[(size_t)(sampBase + M) * 32 + jcol] = acc[r] + bj;
    }
  }

  // ---- FM reduction: 0.5 * sum_k (s_k^2 - sum e_k^2), reduced over 16 threads/sample ----
  float part = 0.0f;
  #pragma unroll
  for (int it = 0; it < 4; ++it) {
    part += (s4[it].x*s4[it].x - q4[it].x) + (s4[it].y*s4[it].y - q4[it].y)
          + (s4[it].z*s4[it].z - q4[it].z) + (s4[it].w*s4[it].w - q4[it].w);
  }
  part *= 0.5f;
  if (lane16 == 0) part += fm1;
  red[t] = part;
  __syncthreads();
  if (t < 16) {
    float a = 0.0f;
    #pragma unroll
    for (int i = 0; i < 16; ++i) a += red[t * 16 + i];
    fm[sampBase + t] = a;
  }
}

// ---------------------------------------------------------------------------
// K2/K4: per-column batch stats (mean, var) over N=8192. Grid = 32 blocks.
// ---------------------------------------------------------------------------
__global__ __launch_bounds__(256) void k_stats(const float* __restrict__ z,
                                               float* __restrict__ stats)
{
  __shared__ float ss[256], sq[256];
  const int j = blockIdx.x;
  const int t = threadIdx.x;
  float s = 0.0f, q = 0.0f;
  for (int i = t; i < N_SAMP; i += 256) {
    const float v = z[(size_t)i * 32 + j];
    s += v; q += v * v;
  }
  ss[t] = s; sq[t] = q;
  __syncthreads();
  for (int off = 128; off > 0; off >>= 1) {
    if (t < off) { ss[t] += ss[t + off]; sq[t] += sq[t + off]; }
    __syncthreads();
  }
  if (t == 0) {
    const float m = ss[0] * (1.0f / N_SAMP);
    stats[j]      = m;
    stats[32 + j] = sq[0] * (1.0f / N_SAMP) - m * m;   // population var (ddof=0)
  }
}

// ---------------------------------------------------------------------------
// K3: h1 = BN1(z1); z2 = h1 @ l2_w^T + l2_b.  One thread per sample.
// ---------------------------------------------------------------------------
__global__ __launch_bounds__(256) void k3_mlp2(
    const float* __restrict__ z1, const float* __restrict__ st1,
    const float* __restrict__ g1, const float* __restrict__ b1,
    const float* __restrict__ l2w, const float* __restrict__ l2b,
    float* __restrict__ z2)
{
  __shared__ float W[32 * 32];
  __shared__ float sc[32], sh[32];
  const int t = threadIdx.x;
  for (int i = t; i < 1024; i += 256) W[i] = l2w[i];
  if (t < 32) {
    const float m = st1[t], v = st1[32 + t];
    const float s = g1[t] * rsqrtf(v + BN_EPS);
    sc[t] = s; sh[t] = b1[t] - m * s;
  }
  __syncthreads();

  const int n = blockIdx.x * 256 + t;
  float h[32];
  #pragma unroll
  for (int j = 0; j < 32; ++j) h[j] = z1[(size_t)n * 32 + j] * sc[j] + sh[j];
  #pragma unroll
  for (int j2 = 0; j2 < 32; ++j2) {
    float a = l2b[j2];
    #pragma unroll
    for (int j = 0; j < 32; ++j) a += h[j] * W[j2 * 32 + j];
    z2[(size_t)n * 32 + j2] = a;
  }
}

// ---------------------------------------------------------------------------
// K5: out[n] = fm[n] + sum_j BN2(z2)[n,j] + bias
// ---------------------------------------------------------------------------
__global__ __launch_bounds__(256) void k5_out(
    const float* __restrict__ fm, const float* __restrict__ z2,
    const float* __restrict__ st2, const float* __restrict__ g2,
    const float* __restrict__ b2, const float* __restrict__ bias,
    float* __restrict__ out)
{
  __shared__ float sc[32], sh[32];
  const int t = threadIdx.x;
  if (t < 32) {
    const float m = st2[t], v = st2[32 + t];
    const float s = g2[t] * rsqrtf(v + BN_EPS);
    sc[t] = s; sh[t] = b2[t] - m * s;
  }
  __syncthreads();
  const int n = blockIdx.x * 256 + t;
  float a = fm[n] + bias[0];
  #pragma unroll
  for (int j = 0; j < 32; ++j) a += z2[(size_t)n * 32 + j] * sc[j] + sh[j];
  out[n] = a;
}

// ---------------------------------------------------------------------------
extern "C" void kernel_launch(void* const* d_in, const int* in_sizes, int n_in,
                              void* d_out, int out_size, void* d_ws, size_t ws_size,
                              hipStream_t stream)
{
  const int*   Xi   = (const int*)  d_in[0];
  const float* Xv   = (const float*)d_in[1];
  const float* W1   = (const float*)d_in[2];
  const float* W2   = (const float*)d_in[3];
  const float* bias = (const float*)d_in[4];
  const float* l1w  = (const float*)d_in[5];
  const float* l1b  = (const float*)d_in[6];
  const float* bn1g = (const float*)d_in[7];
  const float* bn1b = (const float*)d_in[8];
  const float* l2w  = (const float*)d_in[9];
  const float* l2b  = (const float*)d_in[10];
  const float* bn2g = (const float*)d_in[11];
  const float* bn2b = (const float*)d_in[12];
  float* out = (float*)d_out;

  float* fm  = (float*)d_ws;            // 8192
  float* z1  = fm + N_SAMP;             // 8192*32
  float* z2  = z1 + (size_t)N_SAMP*32;  // 8192*32
  float* st1 = z2 + (size_t)N_SAMP*32;  // 64
  float* st2 = st1 + 64;                // 64

  k1_gather_fm_gemm<<<N_SAMP/16, 256, 0, stream>>>(Xi, Xv, W1, W2, l1w, l1b, z1, fm);
  k_stats<<<32, 256, 0, stream>>>(z1, st1);
  k3_mlp2<<<N_SAMP/256, 256, 0, stream>>>(z1, st1, bn1g, bn1b, l2w, l2b, z2);
  k_stats<<<32, 256, 0, stream>>>(z2, st2);
  k5_out<<<N_SAMP/256, 256, 0, stream>>>(fm, z2, st2, bn2g, bn2b, bias, out);
}